// Decoder_32272384262233
// MI455X (gfx1250) — compile-verified
//
#include <hip/hip_runtime.h>
#include <hip/hip_bf16.h>

typedef __attribute__((ext_vector_type(2))) float v2f;
typedef __attribute__((ext_vector_type(8))) float v8f;

#define EPS_BN 1e-5f
#define EPS_W  1e-8f

// ---------------------------------------------------------------------------
// 3-NN search: one thread per target point, serial scan over S sources.
// xyz layout [B,3,N] (reference layout). Outputs idx[t*3+j], w[t*3+j].
// ---------------------------------------------------------------------------
__global__ __launch_bounds__(256) void knn3_kernel(
    const float* __restrict__ xyz1, const float* __restrict__ xyz2,
    int B, int N, int S, int* __restrict__ idxOut, float* __restrict__ wOut)
{
  int t = blockIdx.x * blockDim.x + threadIdx.x;
  if (t >= B * N) return;
  int b = t / N, n = t - b * N;
  const float* p1 = xyz1 + (size_t)b * 3 * N;
  float px = p1[n], py = p1[N + n], pz = p1[2 * N + n];
  const float* p2 = xyz2 + (size_t)b * 3 * S;
  float d0 = 3.4e38f, d1 = 3.4e38f, d2 = 3.4e38f;
  int i0 = 0, i1 = 0, i2 = 0;
  for (int s = 0; s < S; ++s) {
    float dx = px - p2[s];
    float dy = py - p2[S + s];
    float dz = pz - p2[2 * S + s];
    float d = dx * dx + dy * dy + dz * dz;
    if (d < d0)      { d2 = d1; i2 = i1; d1 = d0; i1 = i0; d0 = d; i0 = s; }
    else if (d < d1) { d2 = d1; i2 = i1; d1 = d;  i1 = s; }
    else if (d < d2) { d2 = d;  i2 = s; }
  }
  float w0 = 1.f / (d0 + EPS_W), w1 = 1.f / (d1 + EPS_W), w2 = 1.f / (d2 + EPS_W);
  float inv = 1.f / (w0 + w1 + w2);
  idxOut[t * 3 + 0] = i0; idxOut[t * 3 + 1] = i1; idxOut[t * 3 + 2] = i2;
  wOut[t * 3 + 0] = w0 * inv; wOut[t * 3 + 1] = w1 * inv; wOut[t * 3 + 2] = w2 * inv;
}

// ---------------------------------------------------------------------------
// Pack [B,C,N] (reference layout) -> rows [rowOff..rowOff+C) of [Ctot][B*N].
// ---------------------------------------------------------------------------
__global__ __launch_bounds__(256) void pack_kernel(
    const float* __restrict__ in, float* __restrict__ out,
    int C, int B, int N, int rowOff)
{
  int P = B * N;
  int t = blockIdx.x * blockDim.x + threadIdx.x;
  if (t >= C * P) return;
  int c = t / P, p = t - c * P;
  int b = p / N, n = p - b * N;
  out[(size_t)(rowOff + c) * P + p] = in[((size_t)b * C + c) * N + n];
}

// ---------------------------------------------------------------------------
// Inverse-distance interpolation: feat2 is [C2][B*S]; writes rows
// [rowOff..rowOff+C2) of x [Ctot][B*N].
// ---------------------------------------------------------------------------
__global__ __launch_bounds__(256) void interp_kernel(
    const float* __restrict__ feat2, const int* __restrict__ idx,
    const float* __restrict__ w, float* __restrict__ out,
    int rowOff, int C2, int B, int N, int S)
{
  int P = B * N;
  int t = blockIdx.x * blockDim.x + threadIdx.x;
  if (t >= C2 * P) return;
  int c = t / P, p = t - c * P;
  int b = p / N;
  const int*   ix = idx + (size_t)p * 3;
  const float* ww = w   + (size_t)p * 3;
  const float* f  = feat2 + (size_t)c * ((size_t)B * S) + (size_t)b * S;
  out[(size_t)(rowOff + c) * P + p] =
      ww[0] * f[ix[0]] + ww[1] * f[ix[1]] + ww[2] * f[ix[2]];
}

// ---------------------------------------------------------------------------
// WMMA f32 GEMM: Y[Cout][P] = W[Cout][Cin] * X[Cin][P] + bias.
// One wave computes a 16x64 output strip (4 column tiles) so each A-matrix
// (weight) load feeds 4 V_WMMA_F32_16X16X4_F32 issues.
// A (16x4):  lane m holds row m; lanes 0-15 K={0,1}, lanes 16-31 K={2,3}.
// B (4x16):  lane holds column lane%16; same K split across lane halves.
// D (16x16): lane holds column lane%16; VGPR v holds row v + 8*(lane/16).
// Requires: Cout%16==0, Cin%4==0, P%64==0. Tile predicate is wave-uniform,
// so EXEC is all-ones for every WMMA.
// ---------------------------------------------------------------------------
__global__ __launch_bounds__(256) void gemm_wmma_kernel(
    const float* __restrict__ W, const float* __restrict__ bias,
    const float* __restrict__ X, float* __restrict__ Y,
    int Cout, int Cin, int P)
{
  int gtid  = blockIdx.x * blockDim.x + threadIdx.x;
  int wave  = gtid >> 5;
  int lane  = threadIdx.x & 31;
  int tilesN = P >> 6;              // 64 columns per wave
  int mTile = wave / tilesN;
  int nTile = wave - mTile * tilesN;
  if (mTile * 16 >= Cout) return;   // wave-uniform exit

  int hi   = lane >> 4;             // 0: lanes 0-15, 1: lanes 16-31
  int l16  = lane & 15;
  int aRow = mTile * 16 + l16;      // A-matrix row (output channel)
  int col0 = (nTile << 6) + l16;    // first column handled by this lane
  int kOff = hi << 1;               // K offset carried by this lane half

  const float* wr = W + (size_t)aRow * Cin + kOff;
  const float* xc = X + col0;

  v8f acc0 = {}, acc1 = {}, acc2 = {}, acc3 = {};
  for (int k = 0; k < Cin; k += 4) {
    v2f a;
    a.x = wr[k];
    a.y = wr[k + 1];
    const float* xk = xc + (size_t)(k + kOff) * P;
    v2f b0, b1, b2, b3;
    b0.x = xk[0];  b0.y = xk[P];
    b1.x = xk[16]; b1.y = xk[P + 16];
    b2.x = xk[32]; b2.y = xk[P + 32];
    b3.x = xk[48]; b3.y = xk[P + 48];
    acc0 = __builtin_amdgcn_wmma_f32_16x16x4_f32(false, a, false, b0, (short)0, acc0, false, false);
    acc1 = __builtin_amdgcn_wmma_f32_16x16x4_f32(false, a, false, b1, (short)0, acc1, false, false);
    acc2 = __builtin_amdgcn_wmma_f32_16x16x4_f32(false, a, false, b2, (short)0, acc2, false, false);
    acc3 = __builtin_amdgcn_wmma_f32_16x16x4_f32(false, a, false, b3, (short)0, acc3, false, false);
  }

  int rowBase = mTile * 16 + (hi << 3);
#pragma unroll
  for (int v = 0; v < 8; ++v) {
    float bi = bias[rowBase + v];
    float* yr = Y + (size_t)(rowBase + v) * P + col0;
    yr[0]  = acc0[v] + bi;
    yr[16] = acc1[v] + bi;
    yr[32] = acc2[v] + bi;
    yr[48] = acc3[v] + bi;
  }
}

// ---------------------------------------------------------------------------
// Per-channel sum / sum-of-squares over Y[C][P] into sums[0..C) / sums[C..2C).
// ---------------------------------------------------------------------------
__global__ __launch_bounds__(256) void colsum_kernel(
    const float* __restrict__ Y, float* __restrict__ sums, int C, int P)
{
  __shared__ float s1[256];
  __shared__ float s2[256];
  int c = blockIdx.y;
  float a = 0.f, a2 = 0.f;
  for (int p = blockIdx.x * blockDim.x + threadIdx.x; p < P;
       p += gridDim.x * blockDim.x) {
    float v = Y[(size_t)c * P + p];
    a += v; a2 += v * v;
  }
  s1[threadIdx.x] = a; s2[threadIdx.x] = a2;
  __syncthreads();
  for (int st = 128; st > 0; st >>= 1) {
    if ((int)threadIdx.x < st) {
      s1[threadIdx.x] += s1[threadIdx.x + st];
      s2[threadIdx.x] += s2[threadIdx.x + st];
    }
    __syncthreads();
  }
  if (threadIdx.x == 0) {
    atomicAdd(&sums[c], s1[0]);
    atomicAdd(&sums[C + c], s2[0]);
  }
}

// ---------------------------------------------------------------------------
// BatchNorm (training-mode batch stats, ddof=0) + ReLU, in place.
// ---------------------------------------------------------------------------
__global__ __launch_bounds__(256) void bn_relu_kernel(
    float* __restrict__ Y, const float* __restrict__ sums,
    const float* __restrict__ gamma, const float* __restrict__ beta,
    int C, int P)
{
  int t = blockIdx.x * blockDim.x + threadIdx.x;
  if (t >= C * P) return;
  int c = t / P;
  float invP = 1.f / (float)P;
  float mean = sums[c] * invP;
  float var  = sums[C + c] * invP - mean * mean;
  float a    = gamma[c] * rsqrtf(var + EPS_BN);
  float sh   = beta[c] - mean * a;
  float v    = fmaf(Y[t], a, sh);
  Y[t] = v > 0.f ? v : 0.f;
}

// ---------------------------------------------------------------------------
// Head: 13x32 conv + log_softmax over channels + transpose to [P][13].
// ---------------------------------------------------------------------------
__global__ __launch_bounds__(256) void head_kernel(
    const float* __restrict__ X, const float* __restrict__ W2,
    const float* __restrict__ b2, float* __restrict__ out, int P)
{
  __shared__ float w[13 * 32];
  __shared__ float bb[13];
  for (int i = threadIdx.x; i < 13 * 32; i += blockDim.x) w[i] = W2[i];
  if (threadIdx.x < 13) bb[threadIdx.x] = b2[threadIdx.x];
  __syncthreads();
  int p = blockIdx.x * blockDim.x + threadIdx.x;
  if (p >= P) return;
  float xv[32];
#pragma unroll
  for (int c = 0; c < 32; ++c) xv[c] = X[(size_t)c * P + p];
  float o[13];
  float mx = -3.4e38f;
#pragma unroll
  for (int k = 0; k < 13; ++k) {
    float s = bb[k];
#pragma unroll
    for (int c = 0; c < 32; ++c) s = fmaf(w[k * 32 + c], xv[c], s);
    o[k] = s;
    mx = fmaxf(mx, s);
  }
  float se = 0.f;
#pragma unroll
  for (int k = 0; k < 13; ++k) se += expf(o[k] - mx);
  float lse = mx + logf(se);
#pragma unroll
  for (int k = 0; k < 13; ++k) out[(size_t)p * 13 + k] = o[k] - lse;
}

// ---------------------------------------------------------------------------
// Host orchestration
// ---------------------------------------------------------------------------
extern "C" void kernel_launch(void* const* d_in, const int* in_sizes, int n_in,
                              void* d_out, int out_size, void* d_ws, size_t ws_size,
                              hipStream_t stream)
{
  (void)out_size; (void)ws_size;
  const int B = 16;

  // ---- input index maps: detect flattening convention from in_sizes[0] ----
  // insertion order: xyz first (16*3*4096 = 196608)
  // jax-pytree alphabetical: f1 first (16*64*1024 = 1048576)
  int I_xyz, I_xyz1, I_f1, I_xyz2, I_f2, I_xyz3, I_f3, I_xyz4, I_f4;
  int I_c1W, I_c1b, I_g1, I_b1, I_c2W, I_c2b;
  int fp4i[2][4], fp3i[2][4], fp2i[2][4], fp1i[3][4];
  bool alpha = (n_in > 0 && in_sizes[0] == 1048576);
  if (!alpha) {
    I_xyz = 0; I_xyz1 = 1; I_f1 = 2; I_xyz2 = 3; I_f2 = 4;
    I_xyz3 = 5; I_f3 = 6; I_xyz4 = 7; I_f4 = 8;
    int k = 9;
    for (int l = 0; l < 2; ++l) for (int j = 0; j < 4; ++j) fp4i[l][j] = k++;
    for (int l = 0; l < 2; ++l) for (int j = 0; j < 4; ++j) fp3i[l][j] = k++;
    for (int l = 0; l < 2; ++l) for (int j = 0; j < 4; ++j) fp2i[l][j] = k++;
    for (int l = 0; l < 3; ++l) for (int j = 0; j < 4; ++j) fp1i[l][j] = k++;
    I_c1W = k++; I_c1b = k++; I_g1 = k++; I_b1 = k++; I_c2W = k++; I_c2b = k++;
  } else {
    I_f1 = 0; I_f2 = 1; I_f3 = 2; I_f4 = 3;
    I_b1 = 4; I_g1 = 5; I_c1W = 6; I_c1b = 7; I_c2W = 8; I_c2b = 9;
    int k = 10;
    for (int l = 0; l < 3; ++l) for (int j = 0; j < 4; ++j) fp1i[l][j] = k++;
    for (int l = 0; l < 2; ++l) for (int j = 0; j < 4; ++j) fp2i[l][j] = k++;
    for (int l = 0; l < 2; ++l) for (int j = 0; j < 4; ++j) fp3i[l][j] = k++;
    for (int l = 0; l < 2; ++l) for (int j = 0; j < 4; ++j) fp4i[l][j] = k++;
    I_xyz = 46; I_xyz1 = 47; I_xyz2 = 48; I_xyz3 = 49; I_xyz4 = 50;
  }
  auto F = [&](int i) { return (const float*)d_in[i]; };

  // ---- workspace carve (floats) ----
  const size_t BUF = 8388608;                 // 128 * 65536 (largest activation)
  float* ws   = (float*)d_ws;
  float* Bf0  = ws;
  float* Bf1  = ws + BUF;
  float* FPK  = ws + 2 * BUF;                 // packed f4: 128*256
  float* sums = FPK + 32768;                  // up to 2*256
  int*   idx  = (int*)(sums + 1024);          // up to 65536*3
  float* wgt  = (float*)(idx + 196608);       // up to 65536*3

  // ---- conv + batchnorm + relu layer ----
  auto conv_bn = [&](int li[4], const float* X, float* Y,
                     int Cout, int Cin, int P) {
    hipMemsetAsync(sums, 0, 2 * (size_t)Cout * sizeof(float), stream);
    long waves  = (long)(Cout / 16) * (P / 64);
    long blocks = (waves * 32 + 255) / 256;
    gemm_wmma_kernel<<<(int)blocks, 256, 0, stream>>>(
        F(li[0]), F(li[1]), X, Y, Cout, Cin, P);
    int gx = (P + 255) / 256; if (gx > 128) gx = 128;
    colsum_kernel<<<dim3(gx, Cout), 256, 0, stream>>>(Y, sums, Cout, P);
    int tot = Cout * P;
    bn_relu_kernel<<<(tot + 255) / 256, 256, 0, stream>>>(
        Y, sums, F(li[2]), F(li[3]), Cout, P);
  };
  auto launch1d = [&](int n) { return (n + 255) / 256; };

  // ================= Stage fp4: xyz3(64) <- xyz4(16) =================
  {
    int N = 64, S = 16, P = B * N;
    knn3_kernel<<<launch1d(B * N), 256, 0, stream>>>(
        F(I_xyz3), F(I_xyz4), B, N, S, idx, wgt);
    pack_kernel<<<launch1d(128 * B * S), 256, 0, stream>>>(F(I_f4), FPK, 128, B, S, 0);
    pack_kernel<<<launch1d(64 * P), 256, 0, stream>>>(F(I_f3), Bf0, 64, B, N, 0);
    interp_kernel<<<launch1d(128 * P), 256, 0, stream>>>(
        FPK, idx, wgt, Bf0, 64, 128, B, N, S);
    conv_bn(fp4i[0], Bf0, Bf1, 256, 192, P);
    conv_bn(fp4i[1], Bf1, Bf0, 256, 256, P);          // OUT_fp4 = Bf0 [256][1024]
  }
  // ================= Stage fp3: xyz2(256) <- xyz3(64) ================
  {
    int N = 256, S = 64, P = B * N;
    knn3_kernel<<<launch1d(B * N), 256, 0, stream>>>(
        F(I_xyz2), F(I_xyz3), B, N, S, idx, wgt);
    pack_kernel<<<launch1d(64 * P), 256, 0, stream>>>(F(I_f2), Bf1, 64, B, N, 0);
    interp_kernel<<<launch1d(256 * P), 256, 0, stream>>>(
        Bf0, idx, wgt, Bf1, 64, 256, B, N, S);
    conv_bn(fp3i[0], Bf1, Bf0, 256, 320, P);
    conv_bn(fp3i[1], Bf0, Bf1, 256, 256, P);          // OUT_fp3 = Bf1 [256][4096]
  }
  // ================= Stage fp2: xyz1(1024) <- xyz2(256) ==============
  {
    int N = 1024, S = 256, P = B * N;
    knn3_kernel<<<launch1d(B * N), 256, 0, stream>>>(
        F(I_xyz1), F(I_xyz2), B, N, S, idx, wgt);
    pack_kernel<<<launch1d(64 * P), 256, 0, stream>>>(F(I_f1), Bf0, 64, B, N, 0);
    interp_kernel<<<launch1d(256 * P), 256, 0, stream>>>(
        Bf1, idx, wgt, Bf0, 64, 256, B, N, S);
    conv_bn(fp2i[0], Bf0, Bf1, 256, 320, P);
    conv_bn(fp2i[1], Bf1, Bf0, 128, 256, P);          // OUT_fp2 = Bf0 [128][16384]
  }
  // ================= Stage fp1: xyz(4096) <- xyz1(1024), no concat ===
  {
    int N = 4096, S = 1024, P = B * N;
    knn3_kernel<<<launch1d(B * N), 256, 0, stream>>>(
        F(I_xyz), F(I_xyz1), B, N, S, idx, wgt);
    interp_kernel<<<launch1d(128 * P), 256, 0, stream>>>(
        Bf0, idx, wgt, Bf1, 0, 128, B, N, S);
    conv_bn(fp1i[0], Bf1, Bf0, 128, 128, P);
    conv_bn(fp1i[1], Bf0, Bf1,  64, 128, P);
    conv_bn(fp1i[2], Bf1, Bf0,  64,  64, P);
    // head: conv1(64->32)+BN+ReLU, conv2(32->13)+log_softmax+transpose
    int c1[4] = { I_c1W, I_c1b, I_g1, I_b1 };
    conv_bn(c1, Bf0, Bf1, 32, 64, P);
    head_kernel<<<launch1d(P), 256, 0, stream>>>(
        Bf1, F(I_c2W), F(I_c2b), (float*)d_out, P);
  }
}